// DeepDFTeqv_expansion_2619930050962
// MI455X (gfx1250) — compile-verified
//
#include <hip/hip_runtime.h>
#include <math.h>

typedef float v2f __attribute__((ext_vector_type(2)));
typedef float v4f __attribute__((ext_vector_type(4)));
typedef float v8f __attribute__((ext_vector_type(8)));

#define NSRC 4096
#define NTGT 2048
#define CUT 4.0f
#define BOHR 0.5291772105638411f
#define EPS 1e-11f

// One wave computes a 16(source) x 16(target) tile.
// dist^2 = |s|^2 - 2 s.t + |t|^2 ; the (|s|^2 - 2 s.t) part comes from a single
// v_wmma_f32_16x16x4_f32 with the K=4 slot carrying (|s|^2, 1).
__global__ __launch_bounds__(256) void deepdft_pairs_kernel(
    const float* __restrict__ src,   // [4096,3] probes  (reference "source")
    const float* __restrict__ tgt,   // [2048,3] atoms   (reference "target")
    float* __restrict__ out)         // dist | direction | mask | expansion
{
  const int lane = threadIdx.x & 31;
  const int wave = threadIdx.x >> 5;
  const int ti   = blockIdx.x >> 4;        // 0..255  i-tile (source rows)
  const int jblk = blockIdx.x & 15;        // 0..15
  const int tj   = jblk * 8 + wave;        // 0..127  j-tile (target cols)
  const int i0 = ti * 16;
  const int j0 = tj * 16;
  const int lo = lane & 15;
  const bool hi = lane >= 16;

  // --- A operand: row i0+lo. lanes0-15 -> K=0,1 ; lanes16-31 -> K=2,3 ---
  const float asx = src[3 * (i0 + lo) + 0];
  const float asy = src[3 * (i0 + lo) + 1];
  const float asz = src[3 * (i0 + lo) + 2];
  const float sn  = asx * asx + asy * asy + asz * asz;

  // --- target column for this lane (also reused for direction/norm) ---
  const float tx = tgt[3 * (j0 + lo) + 0];
  const float ty = tgt[3 * (j0 + lo) + 1];
  const float tz = tgt[3 * (j0 + lo) + 2];
  const float tn = tx * tx + ty * ty + tz * tz;

  v2f A, B;
  A.x = hi ? (-2.0f * asz) : (-2.0f * asx);   // K=2 : K=0
  A.y = hi ? sn            : (-2.0f * asy);   // K=3 : K=1
  B.x = hi ? tz   : tx;                       // K=2 : K=0
  B.y = hi ? 1.0f : ty;                       // K=3 : K=1

  v8f C = {0.f, 0.f, 0.f, 0.f, 0.f, 0.f, 0.f, 0.f};
  // D[r], lane -> (i = i0 + r + 8*hi, j = j0 + (lane&15)) = |s_i|^2 - 2 s_i.t_j
  v8f D = __builtin_amdgcn_wmma_f32_16x16x4_f32(
      /*neg_a=*/false, A, /*neg_b=*/false, B,
      /*c_mod=*/(short)0, C, /*reuse_a=*/false, /*reuse_b=*/false);

  const size_t NM = (size_t)NSRC * NTGT;
  float* out_dist = out;                 // [N,M]
  float* out_dir  = out + NM;            // [N,M,3]
  float* out_mask = out + 4 * NM;        // [N,M]  (bool -> 1.0/0.0)
  float* out_exp  = out + 5 * NM;        // [N,M,20]

  const int j = j0 + lo;
#pragma unroll
  for (int r = 0; r < 8; ++r) {
    const int i = i0 + r + (hi ? 8 : 0);
    const float sx = src[3 * i + 0];     // broadcast across 16 lanes -> cache hit
    const float sy = src[3 * i + 1];
    const float sz = src[3 * i + 2];

    float d2 = D[r] + tn;
    d2 = fmaxf(d2, 0.0f);
    const float dist = __builtin_amdgcn_sqrtf(d2);   // raw v_sqrt_f32
    const size_t pair = (size_t)i * NTGT + j;

    __builtin_nontemporal_store(dist, out_dist + pair);

    const float inv = __builtin_amdgcn_rcpf(dist + EPS);
    __builtin_nontemporal_store((tx - sx) * inv, out_dir + pair * 3 + 0);
    __builtin_nontemporal_store((ty - sy) * inv, out_dir + pair * 3 + 1);
    __builtin_nontemporal_store((tz - sz) * inv, out_dir + pair * 3 + 2);

    __builtin_nontemporal_store(dist < CUT ? 1.0f : 0.0f, out_mask + pair);

    // sinc expansion: sin(n*pi*d/CUT)/d.
    // v_sin/v_cos take revolutions: sin(pi*d/4) = sin(2*pi * d/8) -> one mul,
    // no range-reduction preamble. The Chebyshev recurrence
    //   s_{k+1} = 2cos(th)*s_k - s_{k-1}
    // is linear, so seeding with s1/d makes every term pre-divided by d.
    const float d    = dist * BOHR + EPS;
    const float invd = __builtin_amdgcn_rcpf(d);
    const float rev  = d * 0.125f;                    // d/(2*CUT) revolutions
    const float s1   = __builtin_amdgcn_sinf(rev);    // raw v_sin_f32
    const float c1   = __builtin_amdgcn_cosf(rev);    // raw v_cos_f32
    const float twoc = 2.0f * c1;

    float e[20];
    float sm1 = 0.0f, sk = s1 * invd;                 // s'_1 = sin(th)/d
#pragma unroll
    for (int f = 0; f < 20; ++f) {
      e[f] = sk;
      const float snext = __builtin_fmaf(twoc, sk, -sm1);
      sm1 = sk;
      sk = snext;
    }

    // pair*20 floats = pair*80 bytes -> 16B aligned: 5x b128 NT stores
    v4f* ep = (v4f*)(out_exp + pair * 20);
#pragma unroll
    for (int q = 0; q < 5; ++q) {
      v4f v = {e[4 * q + 0], e[4 * q + 1], e[4 * q + 2], e[4 * q + 3]};
      __builtin_nontemporal_store(v, ep + q);
    }
  }
}

extern "C" void kernel_launch(void* const* d_in, const int* in_sizes, int n_in,
                              void* d_out, int out_size, void* d_ws, size_t ws_size,
                              hipStream_t stream) {
  const float* src = (const float*)d_in[0];  // source [1,4096,3]
  const float* tgt = (const float*)d_in[1];  // target [1,2048,3]
  float* out = (float*)d_out;
  // 256 i-tiles * 16 blocks-per-i-tile ; 8 waves (tiles) per block
  deepdft_pairs_kernel<<<dim3(4096), dim3(256), 0, stream>>>(src, tgt, out);
}